// UnifiedWeightGating_66614942761278
// MI455X (gfx1250) — compile-verified
//
#include <hip/hip_runtime.h>
#include <math.h>

#define B_DIM   8
#define S_DIM   2048
#define D_DIM   1024
#define H_DIM   4
#define HD_DIM  256
#define ROWS_PER_BLOCK 16
#define PAD_B   16                                // batch padded to WMMA M=16

typedef float v2f __attribute__((ext_vector_type(2)));
typedef float v4f __attribute__((ext_vector_type(4)));
typedef float v8f __attribute__((ext_vector_type(8)));

// ---- workspace layout (float offsets) ----
// Global-mean accumulators are padded to 16 rows; rows 8..15 stay zero so the
// WMMA A-fragment loads need no masking (no exec divergence in the K loop).
#define WS_TEXT_SUM   0                           // PAD_B*D (rows 8..15 zero)
#define WS_IMAGE_SUM  (PAD_B * D_DIM)             // PAD_B*D (rows 8..15 zero)
#define WS_SIM_SUM    (2 * PAD_B * D_DIM)         // B
#define WS_DOT_ACC    (WS_SIM_SUM + B_DIM)        // B
#define WS_FINAL_GATE (WS_DOT_ACC + B_DIM)        // B
#define WS_STATS      32800                       // B*S*4 (mu_t, rstd_t, mu_i, rstd_i), 16B aligned
#define WS_ZERO_COUNT (WS_DOT_ACC + B_DIM)        // zero padded sums + sim + dot acc

// ---------------------------------------------------------------------------
__global__ void zero_ws_kernel(float* __restrict__ ws) {
    int i = blockIdx.x * 256 + threadIdx.x;
    if (i < WS_ZERO_COUNT) ws[i] = 0.0f;
}

// ---------------------------------------------------------------------------
// 4-wide block reduction over 256 threads via LDS; result broadcast to all.
__device__ __forceinline__ void block_reduce4(float v0, float v1, float v2, float v3,
                                              float* red, float out[4]) {
    const int tid = threadIdx.x;
    red[0 * 256 + tid] = v0;
    red[1 * 256 + tid] = v1;
    red[2 * 256 + tid] = v2;
    red[3 * 256 + tid] = v3;
    __syncthreads();
#pragma unroll
    for (int off = 128; off > 0; off >>= 1) {
        if (tid < off) {
            red[0 * 256 + tid] += red[0 * 256 + tid + off];
            red[1 * 256 + tid] += red[1 * 256 + tid + off];
            red[2 * 256 + tid] += red[2 * 256 + tid + off];
            red[3 * 256 + tid] += red[3 * 256 + tid + off];
        }
        __syncthreads();
    }
    out[0] = red[0];
    out[1] = red[256];
    out[2] = red[512];
    out[3] = red[768];
    __syncthreads();
}

// ---------------------------------------------------------------------------
// Pass 1: per-row layernorm stats, cosine-sim accumulation, per-batch mean
// accumulation of normalized reps. One block handles 16 rows of one batch.
__global__ __launch_bounds__(256)
void pass1_kernel(const float* __restrict__ text, const float* __restrict__ image,
                  const float* __restrict__ lntw, const float* __restrict__ lntb,
                  const float* __restrict__ lniw, const float* __restrict__ lnib,
                  float* __restrict__ ws) {
    __shared__ float red[1024];
    const int b     = blockIdx.x;
    const int chunk = blockIdx.y;
    const int tid   = threadIdx.x;
    const int d0    = tid * 4;

    const v4f wt4 = *(const v4f*)(lntw + d0);
    const v4f bt4 = *(const v4f*)(lntb + d0);
    const v4f wi4 = *(const v4f*)(lniw + d0);
    const v4f bi4 = *(const v4f*)(lnib + d0);

    float accT[4] = {0.f, 0.f, 0.f, 0.f};
    float accI[4] = {0.f, 0.f, 0.f, 0.f};
    float simAcc  = 0.0f;

    float* stats = ws + WS_STATS;
    const float invD = 1.0f / (float)D_DIM;

    for (int r = 0; r < ROWS_PER_BLOCK; ++r) {
        const int    s    = chunk * ROWS_PER_BLOCK + r;
        const size_t row  = (size_t)b * S_DIM + s;
        const size_t base = row * D_DIM + d0;

        const v4f t4 = *(const v4f*)(text + base);
        const v4f i4 = *(const v4f*)(image + base);

        float st = 0.f, sst = 0.f, si = 0.f, ssi = 0.f;
#pragma unroll
        for (int k = 0; k < 4; ++k) {
            st  += t4[k];  sst += t4[k] * t4[k];
            si  += i4[k];  ssi += i4[k] * i4[k];
        }
        float tot[4];
        block_reduce4(st, sst, si, ssi, red, tot);

        const float mu_t  = tot[0] * invD;
        const float var_t = fmaxf(tot[1] * invD - mu_t * mu_t, 0.0f);
        const float rs_t  = rsqrtf(var_t + 1e-5f);
        const float mu_i  = tot[2] * invD;
        const float var_i = fmaxf(tot[3] * invD - mu_i * mu_i, 0.0f);
        const float rs_i  = rsqrtf(var_i + 1e-5f);

        float dot = 0.f, nt2 = 0.f, ni2 = 0.f;
#pragma unroll
        for (int k = 0; k < 4; ++k) {
            const float tn = (t4[k] - mu_t) * rs_t * wt4[k] + bt4[k];
            const float in = (i4[k] - mu_i) * rs_i * wi4[k] + bi4[k];
            dot += tn * in;
            nt2 += tn * tn;
            ni2 += in * in;
            accT[k] += tn;
            accI[k] += in;
        }
        float tot2[4];
        block_reduce4(dot, nt2, ni2, 0.0f, red, tot2);

        if (tid == 0) {
            const float ntn = fmaxf(sqrtf(tot2[1]), 1e-6f);
            const float nin = fmaxf(sqrtf(tot2[2]), 1e-6f);
            simAcc += tot2[0] / (ntn * nin);
            const v4f st4 = {mu_t, rs_t, mu_i, rs_i};
            *(v4f*)(stats + row * 4) = st4;
        }
    }

    float* tsum = ws + WS_TEXT_SUM + b * D_DIM + d0;
    float* isum = ws + WS_IMAGE_SUM + b * D_DIM + d0;
#pragma unroll
    for (int k = 0; k < 4; ++k) {
        atomicAdd(tsum + k, accT[k]);
        atomicAdd(isum + k, accI[k]);
    }
    if (tid == 0) atomicAdd(ws + WS_SIM_SUM + b, simAcc);
}

// ---------------------------------------------------------------------------
// Gate GEMMs via V_WMMA_F32_16X16X4_F32 (full f32, matches reference math).
// gate[b,j] = (1/S) * (sum_d Tsum[b,d]*Wa[j,d] + sum_d Isum[b,d]*Wq[j,d])
// A rows 8..15 are zero (pre-zeroed workspace) -> unconditional b64 loads,
// no exec-mask divergence. 1/S scaling folded into the epilogue (linear).
// Each wave owns one 16-wide N tile; 256 K-steps, 2 chained WMMAs per step.
__global__ __launch_bounds__(256)
void gates_wmma_kernel(const float* __restrict__ Wa, const float* __restrict__ Wab,
                       const float* __restrict__ Wq, const float* __restrict__ Wqb,
                       const float* __restrict__ simw, const float* __restrict__ simb,
                       const float* __restrict__ fcw, float* __restrict__ ws) {
    const int tid  = threadIdx.x;
    const int wave = tid >> 5;
    const int lane = tid & 31;
    const int tile = blockIdx.x * 8 + wave;     // 0..63
    const int j0   = tile * 16;
    const int m    = lane & 15;                 // A-matrix row held by this lane
    const int n    = lane & 15;                 // B-matrix column held by this lane
    const int ksel = (lane >> 4) * 2;           // lanes 0-15 -> K0/K1, lanes 16-31 -> K2/K3

    const float invS = 1.0f / (float)S_DIM;
    const float* trow   = ws + WS_TEXT_SUM  + (size_t)m * D_DIM + ksel;
    const float* irow   = ws + WS_IMAGE_SUM + (size_t)m * D_DIM + ksel;
    const float* wa_row = Wa + (size_t)(j0 + n) * D_DIM + ksel;
    const float* wq_row = Wq + (size_t)(j0 + n) * D_DIM + ksel;

    v8f c = {0.f, 0.f, 0.f, 0.f, 0.f, 0.f, 0.f, 0.f};

#pragma unroll 4
    for (int k = 0; k < D_DIM; k += 4) {
        v2f at = *(const v2f*)(trow + k);
        v2f ai = *(const v2f*)(irow + k);
        v2f ba = *(const v2f*)(wa_row + k);
        v2f bq = *(const v2f*)(wq_row + k);
        // D = A*B + C, f32 16x16x4
        c = __builtin_amdgcn_wmma_f32_16x16x4_f32(false, at, false, ba, (short)0, c,
                                                  false, false);
        c = __builtin_amdgcn_wmma_f32_16x16x4_f32(false, ai, false, bq, (short)0, c,
                                                  false, false);
    }

    // C layout: lanes 0-15 hold rows M=0..7 in c[0..7] -> the 8 real batches.
    if (lane < 16) {
        const int   j    = j0 + lane;
        const int   khd  = j & (HD_DIM - 1);
        const float bias = Wab[j] + Wqb[j];
        const float fw   = fcw[j];
        const float sw   = simw[khd];
        const float sb   = simb[khd];
#pragma unroll
        for (int v = 0; v < B_DIM; ++v) {
            const float gsim = ws[WS_SIM_SUM + v] * invS;
            float g = c[v] * invS + bias + gsim * sw + sb;
            g = fmaxf(g, 0.0f);                 // relu
            atomicAdd(ws + WS_DOT_ACC + v, g * fw);
        }
    }
}

// ---------------------------------------------------------------------------
__global__ void finalize_gate_kernel(const float* __restrict__ fcb, float* __restrict__ ws) {
    const int b = threadIdx.x;
    if (b < B_DIM) {
        const float x = ws[WS_DOT_ACC + b] + fcb[0];
        ws[WS_FINAL_GATE + b] = 1.0f / (1.0f + expf(-x));
    }
}

// ---------------------------------------------------------------------------
// Pass 2: re-read inputs (served from 192MB L2), renormalize using stored
// stats, blend with per-batch gate. Output stores are non-temporal so the
// write-once 64MB stream does not evict the L2-resident inputs mid-pass.
__global__ __launch_bounds__(256)
void blend_kernel(const float* __restrict__ text, const float* __restrict__ image,
                  const float* __restrict__ lntw, const float* __restrict__ lntb,
                  const float* __restrict__ lniw, const float* __restrict__ lnib,
                  const float* __restrict__ ws, float* __restrict__ out) {
    const size_t row = blockIdx.x;
    const int    b   = (int)(row >> 11);        // row / S_DIM
    const int    tid = threadIdx.x;
    const int    d0  = tid * 4;

    const v4f st = *(const v4f*)(ws + WS_STATS + row * 4);
    const float mu_t = st[0], rs_t = st[1], mu_i = st[2], rs_i = st[3];
    const float g  = ws[WS_FINAL_GATE + b];
    const float gi = 1.0f - g;

    const size_t base = row * D_DIM + d0;
    const v4f t4 = *(const v4f*)(text + base);
    const v4f i4 = *(const v4f*)(image + base);
    const v4f wt = *(const v4f*)(lntw + d0);
    const v4f bt = *(const v4f*)(lntb + d0);
    const v4f wi = *(const v4f*)(lniw + d0);
    const v4f bi = *(const v4f*)(lnib + d0);

    v4f o;
#pragma unroll
    for (int k = 0; k < 4; ++k) {
        const float tn = (t4[k] - mu_t) * rs_t * wt[k] + bt[k];
        const float in = (i4[k] - mu_i) * rs_i * wi[k] + bi[k];
        o[k] = g * tn + gi * in;
    }

    __builtin_nontemporal_store(o, (v4f*)(out + base));
}

// ---------------------------------------------------------------------------
extern "C" void kernel_launch(void* const* d_in, const int* in_sizes, int n_in,
                              void* d_out, int out_size, void* d_ws, size_t ws_size,
                              hipStream_t stream) {
    const float* text  = (const float*)d_in[0];
    const float* image = (const float*)d_in[1];
    const float* lntw  = (const float*)d_in[2];
    const float* lntb  = (const float*)d_in[3];
    const float* lniw  = (const float*)d_in[4];
    const float* lnib  = (const float*)d_in[5];
    const float* Wa    = (const float*)d_in[6];
    const float* Wab   = (const float*)d_in[7];
    const float* Wq    = (const float*)d_in[8];
    const float* Wqb   = (const float*)d_in[9];
    const float* simw  = (const float*)d_in[10];
    const float* simb  = (const float*)d_in[11];
    const float* fcw   = (const float*)d_in[12];
    const float* fcb   = (const float*)d_in[13];
    float* ws  = (float*)d_ws;
    float* out = (float*)d_out;

    zero_ws_kernel<<<(WS_ZERO_COUNT + 255) / 256, 256, 0, stream>>>(ws);

    pass1_kernel<<<dim3(B_DIM, S_DIM / ROWS_PER_BLOCK), 256, 0, stream>>>(
        text, image, lntw, lntb, lniw, lnib, ws);

    // 64 tiles of 16 output columns -> 8 blocks * 8 waves
    gates_wmma_kernel<<<8, 256, 0, stream>>>(Wa, Wab, Wq, Wqb, simw, simb, fcw, ws);

    finalize_gate_kernel<<<1, 32, 0, stream>>>(fcb, ws);

    blend_kernel<<<B_DIM * S_DIM, 256, 0, stream>>>(
        text, image, lntw, lntb, lniw, lnib, ws, out);
}